// RoIHeads_59923383714455
// MI455X (gfx1250) — compile-verified
//
#include <hip/hip_runtime.h>
#include <math.h>

// ---------------- problem constants (from reference) ----------------
#define NB    16          // batch
#define PN    16384       // proposals per image
#define NG    128         // gt boxes per image
#define NM    (PN + NG)   // 16512 = 129 * 128  (divides tile size exactly)
#define NPOS  128
#define NNEG  384
#define NSAMP 512

typedef __attribute__((ext_vector_type(2))) float v2f;
typedef __attribute__((ext_vector_type(8))) float v8f;

// ---------------- JAX threefry2x32 (bit-exact) ----------------
__device__ __forceinline__ unsigned rotl32(unsigned x, int d) {
  return (x << d) | (x >> (32 - d));
}

__device__ __forceinline__ void threefry2x32(unsigned k0, unsigned k1,
                                             unsigned x0, unsigned x1,
                                             unsigned& o0, unsigned& o1) {
  unsigned ks2 = 0x1BD11BDAu ^ k0 ^ k1;
  x0 += k0; x1 += k1;
#define QR(a) { x0 += x1; x1 = rotl32(x1, a); x1 ^= x0; }
  QR(13) QR(15) QR(26) QR(6)  x0 += k1;  x1 += ks2 + 1u;
  QR(17) QR(29) QR(16) QR(24) x0 += ks2; x1 += k0 + 2u;
  QR(13) QR(15) QR(26) QR(6)  x0 += k0;  x1 += k1 + 3u;
  QR(17) QR(29) QR(16) QR(24) x0 += k1;  x1 += ks2 + 4u;
  QR(13) QR(15) QR(26) QR(6)
#undef QR
  o0 = x0 + ks2;
  o1 = x1 + k0 + 5u;
}

// rscores = jax.random.uniform(jax.random.key(42), (B, M)); element j (row-major)
__device__ __forceinline__ float jax_uniform_elem(unsigned j) {
  const unsigned n = (unsigned)NB * (unsigned)NM;   // 264192 (even)
  const unsigned half = n >> 1;
  unsigned c0, c1;
  if (j < half) { c0 = j; c1 = j + half; } else { c0 = j - half; c1 = j; }
  unsigned o0, o1;
  threefry2x32(0u, 42u, c0, c1, o0, o1);            // key(42) -> (hi=0, lo=42)
  unsigned bits = (j < half) ? o0 : o1;
  float f = __uint_as_float((bits >> 9) | 0x3f800000u) - 1.0f;
  return f < 0.0f ? 0.0f : f;                       // lax.max(minval, ...)
}

// monotone float->uint mapping (larger float -> larger uint)
__device__ __forceinline__ unsigned ord_f32(float f) {
  unsigned b = __float_as_uint(f);
  return b ^ ((b >> 31) ? 0xFFFFFFFFu : 0x80000000u);
}

// =====================================================================
// Kernel A: match proposals to GTs.
//   grid = (NM/128, NB), block = 256 (8 waves; each wave owns 16 columns).
//   WMMA f32 16x16x4 computes the 16(gt) x 16(prop) tile of
//   area_g + area_b as a rank-2 outer product per GT chunk.
// =====================================================================
__global__ __launch_bounds__(256)
void roih_match_kernel(const float* __restrict__ proposals,   // [B,PN,4]
                       const float* __restrict__ gt_boxes,    // [B,NG,4]
                       const int*   __restrict__ gt_labels,   // [B,NG]
                       unsigned long long* __restrict__ posKey,  // [B,NM]
                       unsigned long long* __restrict__ negKey,  // [B,NM]
                       int* __restrict__ labArr,                 // [B,NM]
                       int* __restrict__ clampArr)               // [B,NM]
{
  __shared__ float sGT[NG * 4];
  __shared__ float sAreaG[NG];
  __shared__ int   sLab[NG];

  const int img  = blockIdx.y;
  const int tile = blockIdx.x;          // 129 tiles of 128 columns
  const int tid  = threadIdx.x;
  const int lane = tid & 31;
  const int wave = tid >> 5;

  // ---- async global->LDS staging of the GT boxes (gfx1250 async path) ----
  if (tid < NG) {
    const float* gsrc = gt_boxes + ((size_t)img * NG + tid) * 4;
    unsigned ldsOff = (unsigned)(unsigned long long)(const void*)&sGT[tid * 4];
    asm volatile("global_load_async_to_lds_b128 %0, %1, off"
                 :: "v"(ldsOff), "v"(gsrc) : "memory");
    sLab[tid] = gt_labels[(size_t)img * NG + tid];
  }
  asm volatile("s_wait_asynccnt 0" ::: "memory");
  __syncthreads();
  if (tid < NG) {
    float x1 = sGT[tid*4+0], y1 = sGT[tid*4+1];
    float x2 = sGT[tid*4+2], y2 = sGT[tid*4+3];
    sAreaG[tid] = (x2 - x1) * (y2 - y1);
  }
  __syncthreads();

  const int  colm = lane & 15;
  const int  col  = tile * 128 + wave * 16 + colm;   // < NM always (exact tiling)
  const bool lo   = (lane < 16);

  // this column's (appended) proposal box
  float p0, p1, p2, p3;
  if (col < PN) {
    const float4 pb = *(const float4*)(proposals + ((size_t)img * PN + col) * 4);
    p0 = pb.x; p1 = pb.y; p2 = pb.z; p3 = pb.w;
  } else {
    const int g = col - PN;
    p0 = sGT[g*4+0]; p1 = sGT[g*4+1]; p2 = sGT[g*4+2]; p3 = sGT[g*4+3];
  }
  const float areaB = (p2 - p0) * (p3 - p1);

  // B fragment (4x16): row K0 = 1, row K1 = area_b[col]; K2/K3 = 0 (lanes 16-31)
  v2f bfrag;
  bfrag.x = lo ? 1.0f  : 0.0f;
  bfrag.y = lo ? areaB : 0.0f;

  float bestIou = -1.0f;
  int   bestG   = 0;

  for (int t = 0; t < 8; ++t) {                       // 8 chunks of 16 GTs
    // A fragment (16x4): col K0 = area_g[row], col K1 = 1; K2/K3 = 0
    v2f afrag;
    afrag.x = lo ? sAreaG[t * 16 + colm] : 0.0f;
    afrag.y = lo ? 1.0f : 0.0f;
    v8f czero = {};
    // D[g][m] = area_g[g] + area_b[m]  (exact rank-2 outer product)
    v8f den = __builtin_amdgcn_wmma_f32_16x16x4_f32(
        false, afrag, false, bfrag, (short)0, czero, false, false);

#pragma unroll
    for (int r = 0; r < 8; ++r) {
      const int g = t * 16 + r + (lo ? 0 : 8);        // row carried by this half
      const float gx1 = sGT[g*4+0], gy1 = sGT[g*4+1];
      const float gx2 = sGT[g*4+2], gy2 = sGT[g*4+3];
      const float lx = fmaxf(gx1, p0), ly = fmaxf(gy1, p1);
      const float rx = fminf(gx2, p2), ry = fminf(gy2, p3);
      const float w = fmaxf(rx - lx, 0.0f);
      const float h = fmaxf(ry - ly, 0.0f);
      const float inter = w * h;
      const float iou = inter / (den[r] - inter);     // IEEE div: matches ref
      if (iou > bestIou) { bestIou = iou; bestG = g; }  // first-max wins per lane
    }
  }

  // combine the two half-row sets (lane m holds g in {16t..16t+7}, lane m+16 the rest)
  {
    const float oI = __shfl_xor(bestIou, 16, 32);
    const int   oG = __shfl_xor(bestG,   16, 32);
    if (oI > bestIou || (oI == bestIou && oG < bestG)) { bestIou = oI; bestG = oG; }
  }

  // thresholds: FG==BG==0.5 so the "-2 / ignore" branch is dead
  int label, clamped;
  if (bestIou < 0.5f) { label = 0;            clamped = 0;     }
  else                { label = sLab[bestG];  clamped = bestG; }

  if (lo) {
    const float r  = jax_uniform_elem((unsigned)(img * NM + col));
    const float pk = (label >= 1) ? r : -1.0f;
    const float nk = (label == 0) ? r : -1.0f;
    const size_t o = (size_t)img * NM + col;
    // key = (ordered value << 32) | ~index  => u64 max == top_k order
    posKey[o] = ((unsigned long long)ord_f32(pk) << 32) | (0xFFFFFFFFu - (unsigned)col);
    negKey[o] = ((unsigned long long)ord_f32(nk) << 32) | (0xFFFFFFFFu - (unsigned)col);
    labArr[o]   = label;
    clampArr[o] = clamped;
  }
}

// =====================================================================
// Kernel B: per-image top-k sampling + box encoding.
//   grid = NB, block = 1024 (32 wave32s). Hierarchical argmax:
//   per-thread slice maxima are cached in LDS; each of the 512 passes
//   reduces 1024 LDS entries, and only the winner's owner thread
//   rescans its 17-element slice (one element was zeroed).
// =====================================================================
__device__ __forceinline__ unsigned long long shflxor64(unsigned long long v, int m) {
  unsigned lov = (unsigned)v, hiv = (unsigned)(v >> 32);
  lov = __shfl_xor(lov, m, 32);
  hiv = __shfl_xor(hiv, m, 32);
  return ((unsigned long long)hiv << 32) | lov;
}

__global__ __launch_bounds__(1024)
void roih_select_kernel(const float* __restrict__ proposals,  // [B,PN,4]
                        const float* __restrict__ gt_boxes,   // [B,NG,4]
                        unsigned long long* __restrict__ posKey,
                        unsigned long long* __restrict__ negKey,
                        const int* __restrict__ labArr,
                        const int* __restrict__ clampArr,
                        float* __restrict__ outReg,           // [B,512,4]
                        float* __restrict__ outLab,           // [B,512]
                        float* __restrict__ outIdx)           // [B,512]
{
  __shared__ unsigned long long sChunk[1024];   // per-thread slice maxima
  __shared__ unsigned long long sRed[32];       // per-wave partials
  __shared__ int sSamp[NSAMP];
  __shared__ int sSel;

  const int img  = blockIdx.x;
  const int tid  = threadIdx.x;
  const int lane = tid & 31;
  const int wave = tid >> 5;

  for (int phase = 0; phase < 2; ++phase) {
    unsigned long long* K = (phase ? negKey : posKey) + (size_t)img * NM;
    const int cnt     = phase ? NNEG : NPOS;
    const int outBase = phase ? NPOS : 0;

    // build per-thread slice maxima (slice = {tid, tid+1024, ...})
    {
      unsigned long long m = 0ull;
      for (int idx = tid; idx < NM; idx += 1024) {
        const unsigned long long v = K[idx];
        if (v > m) m = v;
      }
      sChunk[tid] = m;
    }
    __syncthreads();

    for (int k = 0; k < cnt; ++k) {
      // reduce 1024 cached maxima (1 LDS read + wave32 shfl tree each)
      unsigned long long best = sChunk[tid];
      for (int m = 16; m >= 1; m >>= 1) {
        const unsigned long long o = shflxor64(best, m);
        if (o > best) best = o;
      }
      if (lane == 0) sRed[wave] = best;
      __syncthreads();
      if (tid == 0) {
        unsigned long long b = sRed[0];
        for (int wv = 1; wv < 32; ++wv) if (sRed[wv] > b) b = sRed[wv];
        const int sel = (int)(0xFFFFFFFFu - (unsigned)(b & 0xFFFFFFFFull));
        sSamp[outBase + k] = sel;
        sSel = sel;
        K[sel] = 0ull;                  // remove; visible after __syncthreads
      }
      __syncthreads();
      // only the owner of the winning slice rescans it (17 L2 reads)
      const int owner = sSel & 1023;
      if (tid == owner) {
        unsigned long long m = 0ull;
        for (int idx = tid; idx < NM; idx += 1024) {
          const unsigned long long v = K[idx];
          if (v > m) m = v;
        }
        sChunk[tid] = m;
      }
      __syncthreads();
    }
  }

  // gather + encode the 512 sampled rows
  for (int k = tid; k < NSAMP; k += 1024) {
    const int s = sSamp[k];
    const size_t o = (size_t)img * NM + s;
    const int label = labArr[o];
    const int cl    = clampArr[o];

    float p0, p1, p2, p3;
    if (s < PN) {
      const float4 pb = *(const float4*)(proposals + ((size_t)img * PN + s) * 4);
      p0 = pb.x; p1 = pb.y; p2 = pb.z; p3 = pb.w;
    } else {
      const float4 gb = *(const float4*)(gt_boxes + ((size_t)img * NG + (s - PN)) * 4);
      p0 = gb.x; p1 = gb.y; p2 = gb.z; p3 = gb.w;
    }
    const float4 g = *(const float4*)(gt_boxes + ((size_t)img * NG + cl) * 4);

    const float rw  = p2 - p0,  rh  = p3 - p1;
    const float rcx = p0 + 0.5f * rw, rcy = p1 + 0.5f * rh;
    const float gw  = g.z - g.x, gh  = g.w - g.y;
    const float gcx = g.x + 0.5f * gw, gcy = g.y + 0.5f * gh;

    float4 enc;
    enc.x = 10.0f * (gcx - rcx) / rw;
    enc.y = 10.0f * (gcy - rcy) / rh;
    enc.z = 5.0f * logf(gw / rw);
    enc.w = 5.0f * logf(gh / rh);

    *(float4*)(outReg + ((size_t)img * NSAMP + k) * 4) = enc;
    outLab[(size_t)img * NSAMP + k] = (float)label;
    outIdx[(size_t)img * NSAMP + k] = (float)s;
  }
}

// =====================================================================
extern "C" void kernel_launch(void* const* d_in, const int* in_sizes, int n_in,
                              void* d_out, int out_size, void* d_ws, size_t ws_size,
                              hipStream_t stream) {
  (void)in_sizes; (void)n_in; (void)out_size; (void)ws_size;

  const float* proposals = (const float*)d_in[0];   // [16,16384,4] f32
  const float* gt_boxes  = (const float*)d_in[1];   // [16,128,4]   f32
  const int*   gt_labels = (const int*)  d_in[2];   // [16,128]     i32

  // workspace: posKey, negKey (u64 each), labels, clamped (i32 each) = ~6.4 MB
  unsigned long long* posKey = (unsigned long long*)d_ws;
  unsigned long long* negKey = posKey + (size_t)NB * NM;
  int* labArr   = (int*)(negKey + (size_t)NB * NM);
  int* clampArr = labArr + (size_t)NB * NM;

  float* out    = (float*)d_out;
  float* outReg = out;                               // 16*512*4
  float* outLab = out + (size_t)NB * NSAMP * 4;      // 16*512
  float* outIdx = outLab + (size_t)NB * NSAMP;       // 16*512

  dim3 gA(NM / 128, NB);                             // 129 x 16 blocks
  roih_match_kernel<<<gA, 256, 0, stream>>>(proposals, gt_boxes, gt_labels,
                                            posKey, negKey, labArr, clampArr);

  roih_select_kernel<<<NB, 1024, 0, stream>>>(proposals, gt_boxes,
                                              posKey, negKey, labArr, clampArr,
                                              outReg, outLab, outIdx);
}